// BitGatConv_48524540510800
// MI455X (gfx1250) — compile-verified
//
#include <hip/hip_runtime.h>
#include <math.h>

#define N_NODES_C 50000
#define N_EDGES_C 800000
#define IN_CH_C   128
#define HC_C      64
#define NEG_SLOPE_C 0.2f
#define M_TILES_C (N_NODES_C / 16)              // 3125 exactly
#define NHC_C ((size_t)N_NODES_C * HC_C)        // 3,200,000 elements

// LDS fragment-store geometry: float2 pair {M[k][n], M[k+1][n]} for even k.
// index = (k>>2)*KSTEP_STRIDE + ((k>>1)&1)*KHALF_STRIDE + n
// KHALF_STRIDE = 80 ==> 80 mod 32 == 16, so lanes 0-15 (khalf 0) and lanes
// 16-31 (khalf 1) occupy disjoint bank-pair sets -> conflict-free ds_load_b64.
#define KHALF_STRIDE 80
#define KSTEP_STRIDE 160

typedef __attribute__((ext_vector_type(2))) float v2f;
typedef __attribute__((ext_vector_type(8))) float v8f;

// ---- order-preserving float<->uint key for atomic segment-max ------------
__device__ __forceinline__ unsigned f32_key(float f) {
  unsigned b = __float_as_uint(f);
  return (b & 0x80000000u) ? ~b : (b | 0x80000000u);
}
__device__ __forceinline__ float key_f32(unsigned k) {
  unsigned b = (k & 0x80000000u) ? (k ^ 0x80000000u) : ~k;
  return __uint_as_float(b);
}
__device__ __forceinline__ float lrelu(float x) {
  return x > 0.0f ? x : x * NEG_SLOPE_C;
}

// ---- init: out = bias, denom = 0, segmax = key(-inf) ---------------------
__global__ __launch_bounds__(256) void gat_init(float* __restrict__ out,
                                                const float* __restrict__ bias,
                                                float* __restrict__ den,
                                                unsigned* __restrict__ smax) {
  size_t i = (size_t)blockIdx.x * blockDim.x + threadIdx.x;
  if (i >= NHC_C) return;
  out[i]  = bias[i & (HC_C - 1)];
  den[i]  = 0.0f;
  smax[i] = f32_key(-INFINITY);   // == 0x007FFFFF
}

// ---- H = X(50000x128) @ W(128x64) via V_WMMA_F32_16X16X4_F32 -------------
// Block: 128 threads = 4 waves, 2 M-tiles per wave (8 tiles/block).
// W staged once per block into LDS in fragment order (ds_load_b64 B frags).
__global__ __launch_bounds__(128) void gat_gemm_xw(const float* __restrict__ X,
                                                   const float* __restrict__ W,
                                                   float* __restrict__ H) {
  __shared__ v2f ldsW[32 * KSTEP_STRIDE];   // 40 KB

  // stage W: 4096 float2 pairs, 32 per thread, coalesced global reads
#pragma unroll 4
  for (int it = 0; it < 32; ++it) {
    int id = it * 128 + threadIdx.x;        // 0..4095
    int n  = id & 63;
    int k2 = id >> 6;                       // 0..63 ; k = 2*k2
    v2f p;
    p.x = W[(size_t)(2 * k2) * HC_C + n];
    p.y = W[(size_t)(2 * k2 + 1) * HC_C + n];
    ldsW[(k2 >> 1) * KSTEP_STRIDE + (k2 & 1) * KHALF_STRIDE + n] = p;
  }
  __syncthreads();

  const int lane  = threadIdx.x & 31;
  const int wave  = threadIdx.x >> 5;
  const int t0    = blockIdx.x * 8 + wave * 2;      // first M-tile
  const int t1    = t0 + 1;
  const int khalf = (lane >> 4) << 1;               // 0 or 2
  const int ncol  = lane & 15;

  // clamp rows so tail waves issue valid addresses with full EXEC
  const int m0 = min(t0 * 16 + (lane & 15), N_NODES_C - 1);
  const int m1 = min(t1 * 16 + (lane & 15), N_NODES_C - 1);
  const float* xr0 = X + (size_t)m0 * IN_CH_C;
  const float* xr1 = X + (size_t)m1 * IN_CH_C;
  const v2f* wfrag = ldsW + (lane >> 4) * KHALF_STRIDE + ncol;

  v8f acc0[4] = {v8f{}, v8f{}, v8f{}, v8f{}};
  v8f acc1[4] = {v8f{}, v8f{}, v8f{}, v8f{}};

  for (int k0 = 0; k0 < IN_CH_C; k0 += 4) {
    v2f a0 = *(const v2f*)(xr0 + k0 + khalf);
    v2f a1 = *(const v2f*)(xr1 + k0 + khalf);
    const v2f* wk = wfrag + (k0 >> 2) * KSTEP_STRIDE;
#pragma unroll
    for (int nt = 0; nt < 4; ++nt) {
      v2f b = wk[nt * 16];
      acc0[nt] = __builtin_amdgcn_wmma_f32_16x16x4_f32(
          false, a0, false, b, (short)0, acc0[nt], false, false);
      acc1[nt] = __builtin_amdgcn_wmma_f32_16x16x4_f32(
          false, a1, false, b, (short)0, acc1[nt], false, false);
    }
  }

  // C/D layout: VGPR v, lanes 0-15 -> M=v, lanes 16-31 -> M=v+8
  const int roff = ((lane >> 4) << 3);
  if (t0 < M_TILES_C) {
#pragma unroll
    for (int v = 0; v < 8; ++v) {
      float* hr = H + (size_t)(t0 * 16 + roff + v) * HC_C + ncol;
#pragma unroll
      for (int nt = 0; nt < 4; ++nt) hr[nt * 16] = acc0[nt][v];
    }
  }
  if (t1 < M_TILES_C) {
#pragma unroll
    for (int v = 0; v < 8; ++v) {
      float* hr = H + (size_t)(t1 * 16 + roff + v) * HC_C + ncol;
#pragma unroll
      for (int nt = 0; nt < 4; ++nt) hr[nt * 16] = acc1[nt][v];
    }
  }
}

// ---- ATI = H @ A1, ATJ = H @ A2 (fused, shared A frags, LDS B frags) -----
__global__ __launch_bounds__(128) void gat_gemm_att(const float* __restrict__ H,
                                                    const float* __restrict__ A1,
                                                    const float* __restrict__ A2,
                                                    float* __restrict__ ATI,
                                                    float* __restrict__ ATJ) {
  __shared__ v2f ldsA1[16 * KSTEP_STRIDE];  // 20 KB
  __shared__ v2f ldsA2[16 * KSTEP_STRIDE];  // 20 KB

  // stage A1+A2: 2048 pairs each
#pragma unroll 4
  for (int it = 0; it < 16; ++it) {
    int id = it * 128 + threadIdx.x;        // 0..2047
    int n  = id & 63;
    int k2 = id >> 6;                       // 0..31 ; k = 2*k2
    v2f p1, p2;
    p1.x = A1[(size_t)(2 * k2) * HC_C + n];
    p1.y = A1[(size_t)(2 * k2 + 1) * HC_C + n];
    p2.x = A2[(size_t)(2 * k2) * HC_C + n];
    p2.y = A2[(size_t)(2 * k2 + 1) * HC_C + n];
    int idx = (k2 >> 1) * KSTEP_STRIDE + (k2 & 1) * KHALF_STRIDE + n;
    ldsA1[idx] = p1;
    ldsA2[idx] = p2;
  }
  __syncthreads();

  const int lane   = threadIdx.x & 31;
  const int tile_m = blockIdx.x * 4 + (threadIdx.x >> 5);
  if (tile_m >= M_TILES_C) return;          // wave-uniform, after the barrier
  const int khalf = (lane >> 4) << 1;
  const int ncol  = lane & 15;
  const int mrow  = tile_m * 16 + (lane & 15);

  const float* hr = H + (size_t)mrow * HC_C;
  const v2f* f1 = ldsA1 + (lane >> 4) * KHALF_STRIDE + ncol;
  const v2f* f2 = ldsA2 + (lane >> 4) * KHALF_STRIDE + ncol;

  v8f accI[4] = {v8f{}, v8f{}, v8f{}, v8f{}};
  v8f accJ[4] = {v8f{}, v8f{}, v8f{}, v8f{}};

  for (int k0 = 0; k0 < HC_C; k0 += 4) {
    v2f a = *(const v2f*)(hr + k0 + khalf);
    const v2f* p1 = f1 + (k0 >> 2) * KSTEP_STRIDE;
    const v2f* p2 = f2 + (k0 >> 2) * KSTEP_STRIDE;
#pragma unroll
    for (int nt = 0; nt < 4; ++nt) {
      v2f b1 = p1[nt * 16];
      v2f b2 = p2[nt * 16];
      accI[nt] = __builtin_amdgcn_wmma_f32_16x16x4_f32(
          false, a, false, b1, (short)0, accI[nt], false, false);
      accJ[nt] = __builtin_amdgcn_wmma_f32_16x16x4_f32(
          false, a, false, b2, (short)0, accJ[nt], false, false);
    }
  }

  const int rbase = tile_m * 16 + ((lane >> 4) << 3);
#pragma unroll
  for (int v = 0; v < 8; ++v) {
    float* pi = ATI + (size_t)(rbase + v) * HC_C + ncol;
    float* pj = ATJ + (size_t)(rbase + v) * HC_C + ncol;
#pragma unroll
    for (int nt = 0; nt < 4; ++nt) {
      pi[nt * 16] = accI[nt][v];
      pj[nt * 16] = accJ[nt][v];
    }
  }
}

// ---- edge pass 1: per-(tgt,channel) running max of leaky-relu logits -----
__global__ __launch_bounds__(256) void gat_edge_max(const long long* __restrict__ tg,
                                                    const long long* __restrict__ sr,
                                                    const float* __restrict__ ATI,
                                                    const float* __restrict__ ATJ,
                                                    unsigned* __restrict__ smax) {
  int t = blockIdx.x * blockDim.x + threadIdx.x;
  int e = t >> 5;
  if (e >= N_EDGES_C) return;
  int c = (t & 31) << 1;
  long long ti = tg[e], si = sr[e];
  float2 ai = *(const float2*)(ATI + (size_t)ti * HC_C + c);
  float2 aj = *(const float2*)(ATJ + (size_t)si * HC_C + c);
  float lx = lrelu(ai.x + aj.x);
  float ly = lrelu(ai.y + aj.y);
  unsigned* p = smax + (size_t)ti * HC_C + c;
  atomicMax(p,     f32_key(lx));
  atomicMax(p + 1, f32_key(ly));
}

// ---- edge pass 2: denom[tgt,c] += exp(logit - max) -----------------------
__global__ __launch_bounds__(256) void gat_edge_denom(const long long* __restrict__ tg,
                                                      const long long* __restrict__ sr,
                                                      const float* __restrict__ ATI,
                                                      const float* __restrict__ ATJ,
                                                      const unsigned* __restrict__ smax,
                                                      float* __restrict__ den) {
  int t = blockIdx.x * blockDim.x + threadIdx.x;
  int e = t >> 5;
  if (e >= N_EDGES_C) return;
  int c = (t & 31) << 1;
  long long ti = tg[e], si = sr[e];
  float2 ai = *(const float2*)(ATI + (size_t)ti * HC_C + c);
  float2 aj = *(const float2*)(ATJ + (size_t)si * HC_C + c);
  float mx0 = key_f32(smax[(size_t)ti * HC_C + c]);
  float mx1 = key_f32(smax[(size_t)ti * HC_C + c + 1]);
  if (!isfinite(mx0)) mx0 = 0.0f;
  if (!isfinite(mx1)) mx1 = 0.0f;
  float e0 = __expf(lrelu(ai.x + aj.x) - mx0);
  float e1 = __expf(lrelu(ai.y + aj.y) - mx1);
  float* p = den + (size_t)ti * HC_C + c;
  atomicAdd(p,     e0);
  atomicAdd(p + 1, e1);
}

// ---- edge pass 3: out[tgt,c] += h[src,c] * e / denom ---------------------
__global__ __launch_bounds__(256) void gat_edge_out(const long long* __restrict__ tg,
                                                    const long long* __restrict__ sr,
                                                    const float* __restrict__ ATI,
                                                    const float* __restrict__ ATJ,
                                                    const unsigned* __restrict__ smax,
                                                    const float* __restrict__ den,
                                                    const float* __restrict__ H,
                                                    float* __restrict__ out) {
  int t = blockIdx.x * blockDim.x + threadIdx.x;
  int e = t >> 5;
  if (e >= N_EDGES_C) return;
  int c = (t & 31) << 1;
  long long ti = tg[e], si = sr[e];
  float2 ai = *(const float2*)(ATI + (size_t)ti * HC_C + c);
  float2 aj = *(const float2*)(ATJ + (size_t)si * HC_C + c);
  float mx0 = key_f32(smax[(size_t)ti * HC_C + c]);
  float mx1 = key_f32(smax[(size_t)ti * HC_C + c + 1]);
  if (!isfinite(mx0)) mx0 = 0.0f;
  if (!isfinite(mx1)) mx1 = 0.0f;
  float e0 = __expf(lrelu(ai.x + aj.x) - mx0);
  float e1 = __expf(lrelu(ai.y + aj.y) - mx1);
  float d0 = den[(size_t)ti * HC_C + c]     + 1e-16f;
  float d1 = den[(size_t)ti * HC_C + c + 1] + 1e-16f;
  float2 hv = *(const float2*)(H + (size_t)si * HC_C + c);
  float* p = out + (size_t)ti * HC_C + c;
  atomicAdd(p,     hv.x * (e0 / d0));
  atomicAdd(p + 1, hv.y * (e1 / d1));
}

extern "C" void kernel_launch(void* const* d_in, const int* in_sizes, int n_in,
                              void* d_out, int out_size, void* d_ws, size_t ws_size,
                              hipStream_t stream) {
  (void)in_sizes; (void)n_in; (void)out_size; (void)ws_size;
  const float*     X    = (const float*)d_in[0];            // [50000,128]
  const long long* adj  = (const long long*)d_in[1];        // [2,800000] int64
  const float*     W    = (const float*)d_in[2];            // [128,64]
  const float*     A1   = (const float*)d_in[3];            // [64,64]
  const float*     A2   = (const float*)d_in[4];            // [64,64]
  const float*     bias = (const float*)d_in[5];            // [64]
  float* out = (float*)d_out;                               // [50000,64]

  // workspace layout: H | ATI | ATJ | DEN | SMAX  (5 x 12.8 MB = 64 MB)
  float*    H    = (float*)d_ws;
  float*    ATI  = H   + NHC_C;
  float*    ATJ  = ATI + NHC_C;
  float*    DEN  = ATJ + NHC_C;
  unsigned* SMAX = (unsigned*)(DEN + NHC_C);

  const long long* tg = adj;              // adj_list[0] = target/segment
  const long long* sr = adj + N_EDGES_C;  // adj_list[1] = source/gather

  const int initBlocks = (int)((NHC_C + 255) / 256);
  const int xwBlocks   = (M_TILES_C + 7) / 8;                         // 8 tiles/block
  const int attBlocks  = (M_TILES_C + 3) / 4;                         // 4 tiles/block
  const int edgeBlocks = (int)(((size_t)N_EDGES_C * 32 + 255) / 256); // wave per edge

  gat_init<<<initBlocks, 256, 0, stream>>>(out, bias, DEN, SMAX);
  gat_gemm_xw<<<xwBlocks, 128, 0, stream>>>(X, W, H);
  gat_gemm_att<<<attBlocks, 128, 0, stream>>>(H, A1, A2, ATI, ATJ);
  gat_edge_max<<<edgeBlocks, 256, 0, stream>>>(tg, sr, ATI, ATJ, SMAX);
  gat_edge_denom<<<edgeBlocks, 256, 0, stream>>>(tg, sr, ATI, ATJ, SMAX, DEN);
  gat_edge_out<<<edgeBlocks, 256, 0, stream>>>(tg, sr, ATI, ATJ, SMAX, DEN, H, out);
}